// pool_conv_90452011254433
// MI455X (gfx1250) — compile-verified
//
#include <hip/hip_runtime.h>

// Pixel-unshuffle (space-to-depth), k=2, s=2, p=0.
// in : [4, 256, 256, 256] f32   (B, C, H, W)
// out: [4, 1024, 128, 128] f32  (B, 4*C, H/2, W/2)
// out[b, (2i+j)*C + c, ho, wo] = in[b, c, 2*ho+i, 2*wo+j]
//
// Pure permutation: 0 FLOPs, 512 MiB of traffic -> HBM-bound (~23 us at
// 23.3 TB/s). Strategy: one input row (1 KB) per wave32. Stage the row into
// LDS with gfx1250 async b128 copies (ASYNCcnt path, no VGPR load-return),
// s_wait_asynccnt, then deinterleave even/odd columns in VGPRs and write two
// 128-float output rows with non-temporal b128 stores (working set > 192 MB
// L2, every byte touched once -> NT is correct).

typedef float v4f __attribute__((ext_vector_type(4)));

#define THREADS 256          // 8 waves per block
#define ROWS_PER_BLOCK 8     // one row per wave
#define W_IN 256
#define H_IN 256
#define C_IN 256
#define BATCH 4

__global__ __launch_bounds__(THREADS)
void pixel_unshuffle_f32(const float* __restrict__ in, float* __restrict__ out) {
    // 8 rows * 256 floats = 512 x v4f = 8 KB of LDS, one 1 KB row per wave.
    __shared__ v4f sh4[ROWS_PER_BLOCK * 64];

    const int tid  = threadIdx.x;
    const int wave = tid >> 5;            // local row 0..7 (one per wave32)
    const int lane = tid & 31;

    const long rowBase = (long)blockIdx.x * ROWS_PER_BLOCK;  // over B*C*H rows
    const float* grow  = in + (rowBase + wave) * W_IN;       // this wave's row

    // ---- Stage: async copy this wave's 1 KB row into LDS (2 x b128 / lane).
    // Issue k=0 covers bytes [0,512) of the row unit-stride across lanes,
    // k=1 covers [512,1024). Tracked with ASYNCcnt; EXEC is all-ones.
#pragma unroll
    for (int k = 0; k < 2; ++k) {
        const int chunk = wave * 64 + lane + 32 * k;     // 16-byte chunk index
        const float* gsrc = grow + (lane + 32 * k) * 4;  // 16B-aligned
        const unsigned ldsaddr = (unsigned)(size_t)&sh4[chunk]; // low 32b = LDS offset
        asm volatile("global_load_async_to_lds_b128 %0, %1, off th:TH_LOAD_NT"
                     :: "v"(ldsaddr), "v"(gsrc)
                     : "memory");
    }
    // Wave consumes only the row it staged itself -> per-wave wait suffices,
    // no workgroup barrier required.
    asm volatile("s_wait_asynccnt 0x0" ::: "memory");

    // ---- Deinterleave: lane handles input cols w0..w0+7 of its row.
    const v4f a = sh4[wave * 64 + lane * 2 + 0];  // w0+0..3 (ds_load_b128)
    const v4f d = sh4[wave * 64 + lane * 2 + 1];  // w0+4..7
    const v4f ev = {a.x, a.z, d.x, d.z};          // even cols -> j=0 block
    const v4f od = {a.y, a.w, d.y, d.w};          // odd  cols -> j=1 block

    // ---- Output addressing.
    const long gr = rowBase + wave;       // gr = (b*C + c)*H + h
    const int  h  = (int)(gr & (H_IN - 1));
    const int  bc = (int)(gr >> 8);       // b*C + c
    const int  b  = bc >> 8;
    const int  c  = bc & (C_IN - 1);
    const int  i  = h & 1;
    const int  ho = h >> 1;
    const int  wo = lane << 2;            // 4 consecutive output cols per lane

    const long ocE  = (long)b * (4 * C_IN) + (2 * i) * C_IN + c;  // j=0 channel
    const long base = (ocE * 128 + ho) * 128 + wo;
    // j=1 channel is ocE+256 -> flat offset +256*128*128.
    __builtin_nontemporal_store(ev, (v4f*)(out + base));
    __builtin_nontemporal_store(od, (v4f*)(out + base + 256L * 128 * 128));
}

extern "C" void kernel_launch(void* const* d_in, const int* in_sizes, int n_in,
                              void* d_out, int out_size, void* d_ws, size_t ws_size,
                              hipStream_t stream) {
    (void)in_sizes; (void)n_in; (void)d_ws; (void)ws_size; (void)out_size;
    const float* in = (const float*)d_in[0];
    float* out = (float*)d_out;

    const long total_rows = (long)BATCH * C_IN * H_IN;        // 262144
    const int  blocks     = (int)(total_rows / ROWS_PER_BLOCK); // 32768
    pixel_unshuffle_f32<<<blocks, THREADS, 0, stream>>>(in, out);
}